// CrossAttn_18356690223800
// MI455X (gfx1250) — compile-verified
//
#include <hip/hip_runtime.h>
#include <math.h>

typedef __attribute__((ext_vector_type(2))) float v2f;
typedef __attribute__((ext_vector_type(8))) float v8f;

#define EPS_LN 1e-6f
#define EPS_L2 1e-12f

__device__ __forceinline__ float gelu_exact(float v) {
    return 0.5f * v * (1.0f + erff(v * 0.70710678118654752440f));
}

// ---------------------------------------------------------------------------
// Kernel 1: token path.  tn_out[b][n] = gelu(LN(token) @ w_tok + b_tok)
//                                      / max(||.||2, eps) * p[b] * exp(alpha)
// One block per batch row. Tiny cost (~3 MFLOP total).
// ---------------------------------------------------------------------------
__global__ __launch_bounds__(256) void tok_kernel(
    const float* __restrict__ token, const float* __restrict__ p,
    const float* __restrict__ alpha,
    const float* __restrict__ g1, const float* __restrict__ b1,
    const float* __restrict__ w_tok, const float* __restrict__ b_tok,
    float* __restrict__ tn_out)
{
    __shared__ float red[256];
    __shared__ float tnorm[768];
    __shared__ float gbuf[128];
    __shared__ float sm, srstd, snrm;
    const int b = blockIdx.x, t = threadIdx.x;
    const float* tok = token + b * 768;

    // mean over 768
    float s = 0.f;
    for (int i = t; i < 768; i += 256) s += tok[i];
    red[t] = s; __syncthreads();
    for (int off = 128; off > 0; off >>= 1) { if (t < off) red[t] += red[t + off]; __syncthreads(); }
    if (t == 0) sm = red[0] * (1.f / 768.f);
    __syncthreads();
    const float m = sm;

    // variance over 768
    float q = 0.f;
    for (int i = t; i < 768; i += 256) { float d = tok[i] - m; q += d * d; }
    red[t] = q; __syncthreads();
    for (int off = 128; off > 0; off >>= 1) { if (t < off) red[t] += red[t + off]; __syncthreads(); }
    if (t == 0) srstd = rsqrtf(red[0] * (1.f / 768.f) + EPS_LN);
    __syncthreads();
    const float rstd = srstd;

    for (int i = t; i < 768; i += 256) tnorm[i] = (tok[i] - m) * rstd * g1[i] + b1[i];
    __syncthreads();

    // matvec [768] x [768,128] + gelu  (tnorm reads are LDS broadcasts,
    // w_tok reads are coalesced across the 128 active lanes)
    if (t < 128) {
        float acc = b_tok[t];
        for (int k = 0; k < 768; ++k) acc = fmaf(tnorm[k], w_tok[k * 128 + t], acc);
        gbuf[t] = gelu_exact(acc);
    }
    __syncthreads();

    // L2 norm over 128
    float ss = 0.f;
    if (t < 128) { float g = gbuf[t]; ss = g * g; }
    red[t] = ss; __syncthreads();
    for (int off = 128; off > 0; off >>= 1) { if (t < off) red[t] += red[t + off]; __syncthreads(); }
    if (t == 0) snrm = fmaxf(sqrtf(red[0]), EPS_L2);
    __syncthreads();

    if (t < 128) {
        float scale = p[b] * expf(alpha[0]) / snrm;   // fold p * e^alpha into tn
        tn_out[b * 128 + t] = gbuf[t] * scale;
    }
}

// ---------------------------------------------------------------------------
// Kernel 2: fused x path. One block = one batch b, one chunk of 256 rows
// (16 tiles of 16 rows). 8 waves; each wave owns one 16-wide N-tile of the
// [16,192]x[192,128] tile-GEMM via V_WMMA_F32_16X16X4_F32.
//   - B fragments (192x16 per wave) live entirely in registers (48 x v2f),
//     loaded once per block from L2-resident w_x -> inner loop is one
//     ds_load_b64 (A) + one WMMA, with two independent accumulator chains.
//   - x tiles are software double-buffered through registers: commit tile tt
//     to LDS, immediately issue tile tt+1's global_load_b128s, then compute.
//   - LDS ~31 KB -> multiple workgroups per WGP for latency hiding.
//   - LN3(x*attn) reuses LN2 row stats (mean_y = attn*m_x, var_y = attn^2*v_x).
// ---------------------------------------------------------------------------
#define ROWP 196   // padded LDS row stride (floats) for 16x192 tiles

__global__ __launch_bounds__(256) void xattn_kernel(
    const float* __restrict__ x, const float* __restrict__ w_x,
    const float* __restrict__ b_x,
    const float* __restrict__ g2, const float* __restrict__ b2,
    const float* __restrict__ g3, const float* __restrict__ b3,
    const float* __restrict__ tn, float* __restrict__ out)
{
    __shared__ float xs[16 * ROWP];        // raw x tile
    __shared__ float as_[16 * ROWP];       // LN2-normalized A tile
    __shared__ float sg2[192], sb2[192], sg3[192], sb3[192];
    __shared__ float tnl[128];
    __shared__ float mx[16], vx[16];
    __shared__ float pdot[16][8], psq[16][8];
    __shared__ float sattn[16], smeany[16], srstdy[16];

    const int t    = threadIdx.x;
    const int wave = t >> 5;
    const int lane = t & 31;
    const int nl   = lane & 15;            // fragment column / A-row index
    const int hi   = lane >> 4;            // half-wave selector (K sub-pair)
    const int b     = blockIdx.x >> 4;     // batch
    const int chunk = blockIdx.x & 15;     // 256-row chunk within this batch
    const int ncol  = wave * 16 + nl;      // this lane's output column

    for (int i = t; i < 192; i += 256) {
        sg2[i] = g2[i]; sb2[i] = b2[i]; sg3[i] = g3[i]; sb3[i] = b3[i];
    }
    if (t < 128) tnl[t] = tn[b * 128 + t];

    // ---- B fragments into registers, once per block (w_x is L2-resident) ---
    // Lane holds B[4x16] per K-block: b.x = w[k0][ncol], b.y = w[k0+1][ncol],
    // k0 = kb*4 + 2*hi  (ISA 7.12.2 mirrored-A layout).
    v2f bfrag[48];
    #pragma unroll
    for (int kb = 0; kb < 48; ++kb) {
        const int k0 = kb * 4 + hi * 2;
        bfrag[kb].x = w_x[k0 * 128 + ncol];
        bfrag[kb].y = w_x[(k0 + 1) * 128 + ncol];
    }
    const float bxv = b_x[ncol];           // x-proj bias for this lane's column
    __syncthreads();
    const float tnv = tnl[ncol];           // tn value for this lane's column

    // ---- preload tile 0 into registers (3 float4 per thread) ----
    const size_t base = ((size_t)(b * 4096 + chunk * 256)) * 192;
    float4 pf[3];
    {
        const float4* xg = (const float4*)(x + base);
        #pragma unroll
        for (int j = 0; j < 3; ++j) pf[j] = xg[t + j * 256];
    }

    for (int tt = 0; tt < 16; ++tt) {
        const size_t tbase = base + (size_t)tt * 3072;   // 16 rows * 192

        // ---- commit prefetched tile to LDS ----
        #pragma unroll
        for (int j = 0; j < 3; ++j) {
            int e4 = t + j * 256;              // 768 float4 total
            int r = e4 / 48, c4 = e4 - r * 48;
            *(float4*)&xs[r * ROWP + c4 * 4] = pf[j];
        }
        // ---- immediately issue next tile's loads; they fly under compute ---
        if (tt < 15) {
            const float4* xg = (const float4*)(x + tbase + 3072);
            #pragma unroll
            for (int j = 0; j < 3; ++j) pf[j] = xg[t + j * 256];
        }
        __syncthreads();

        // ---- LN2: each wave reduces rows {wave, wave+8} over 192 cols ----
        for (int rr = 0; rr < 2; ++rr) {
            const int r = wave + rr * 8;
            float s = 0.f, q = 0.f;
            for (int j = 0; j < 6; ++j) {
                float xv = xs[r * ROWP + lane + j * 32];
                s += xv; q += xv * xv;
            }
            for (int msk = 16; msk >= 1; msk >>= 1) {
                s += __shfl_xor(s, msk, 32);
                q += __shfl_xor(q, msk, 32);
            }
            float m   = s * (1.f / 192.f);
            float var = q * (1.f / 192.f) - m * m;
            float rst = rsqrtf(var + EPS_LN);
            if (lane == 0) { mx[r] = m; vx[r] = var; }
            for (int j = 0; j < 6; ++j) {
                int c = lane + j * 32;
                as_[r * ROWP + c] = (xs[r * ROWP + c] - m) * rst * sg2[c] + sb2[c];
            }
        }
        __syncthreads();

        // ---- tile GEMM: A from LDS (immediate-offset ds_load_b64),
        //      B from registers, two independent WMMA accumulator chains ----
        v8f acc0 = {}, acc1 = {};
        const float* arow = &as_[nl * ROWP + hi * 2];
        #pragma unroll
        for (int kb = 0; kb < 48; kb += 2) {
            v2f a0 = *(const v2f*)(arow + kb * 4);
            v2f a1 = *(const v2f*)(arow + kb * 4 + 4);
            acc0 = __builtin_amdgcn_wmma_f32_16x16x4_f32(
                false, a0, false, bfrag[kb], (short)0, acc0, false, false);
            acc1 = __builtin_amdgcn_wmma_f32_16x16x4_f32(
                false, a1, false, bfrag[kb + 1], (short)0, acc1, false, false);
        }

        // ---- bias + gelu + per-row {sum g*tn, sum g^2} partials ----
        // C/D layout: VGPR v, lane -> row = v + 8*hi, col = ncol
        #pragma unroll
        for (int v = 0; v < 8; ++v) {
            float g  = gelu_exact(acc0[v] + acc1[v] + bxv);
            float pd = g * tnv;
            float ps = g * g;
            for (int msk = 8; msk >= 1; msk >>= 1) {   // reduce 16-lane groups
                pd += __shfl_xor(pd, msk, 32);
                ps += __shfl_xor(ps, msk, 32);
            }
            if (nl == 0) {
                int r = v + hi * 8;
                pdot[r][wave] = pd;
                psq[r][wave]  = ps;
            }
        }
        __syncthreads();

        // ---- finalize attn + LN3 stats (LN3 of x*attn reuses LN2 stats) ----
        if (t < 16) {
            float dot = 0.f, sq = 0.f;
            for (int wv = 0; wv < 8; ++wv) { dot += pdot[t][wv]; sq += psq[t][wv]; }
            float attn = dot / fmaxf(sqrtf(sq), EPS_L2);
            sattn[t]  = attn;
            smeany[t] = attn * mx[t];
            srstdy[t] = rsqrtf(attn * attn * vx[t] + EPS_LN);
        }
        __syncthreads();

        // ---- out = 0.5*x + (x*attn - mean_y)*rstd_y*g3 + b3 ----
        {
            const int r  = t >> 4;
            const int c0 = (t & 15) * 12;
            const float attn = sattn[r], my = smeany[r], rs = srstdy[r];
            float* og = out + tbase + (size_t)r * 192;
            const float* xr = &xs[r * ROWP];
            #pragma unroll
            for (int c = c0; c < c0 + 12; ++c) {
                float xv = xr[c];
                og[c] = 0.5f * xv + (xv * attn - my) * rs * sg3[c] + sb3[c];
            }
        }
        __syncthreads();   // xs/as_ reused next tile
    }
}

// ---------------------------------------------------------------------------
extern "C" void kernel_launch(void* const* d_in, const int* in_sizes, int n_in,
                              void* d_out, int out_size, void* d_ws, size_t ws_size,
                              hipStream_t stream) {
    const float* x     = (const float*)d_in[0];
    const float* token = (const float*)d_in[1];
    const float* p     = (const float*)d_in[2];
    const float* alpha = (const float*)d_in[3];
    const float* ln1_g = (const float*)d_in[4];
    const float* ln1_b = (const float*)d_in[5];
    const float* w_tok = (const float*)d_in[6];
    const float* b_tok = (const float*)d_in[7];
    const float* ln2_g = (const float*)d_in[8];
    const float* ln2_b = (const float*)d_in[9];
    const float* w_x   = (const float*)d_in[10];
    const float* b_x   = (const float*)d_in[11];
    const float* ln3_g = (const float*)d_in[12];
    const float* ln3_b = (const float*)d_in[13];
    float* out = (float*)d_out;
    float* tn  = (float*)d_ws;                 // 32*128 floats of scratch

    tok_kernel<<<32, 256, 0, stream>>>(token, p, alpha, ln1_g, ln1_b,
                                       w_tok, b_tok, tn);
    xattn_kernel<<<512, 256, 0, stream>>>(x, w_x, b_x, ln2_g, ln2_b,
                                          ln3_g, ln3_b, tn, out);
}